// XMILoss_45887430590815
// MI455X (gfx1250) — compile-verified
//
#include <hip/hip_runtime.h>
#include <hip/hip_bf16.h>
#include <stdint.h>

// ---------------------------------------------------------------------------
// XMILoss for MI455X (gfx1250, wave32).
// Stage 1: 1 wave per feature row (C=256 -> 8 f32/lane). Tile of 8 rows per
//          block staged global->LDS with async b128 loads, rowwise softmax
//          stats reduced across lanes with V_WMMA_F32_16X16X4_F32 (ones-matrix
//          reduction trick) + shfl_xor. NLL gather folded in (lane 0).
// Stage 2: deterministic f64 tree reduction of 8192 block partials.
// ---------------------------------------------------------------------------

#define AS1 __attribute__((address_space(1)))
#define AS3 __attribute__((address_space(3)))

typedef __attribute__((ext_vector_type(2))) float v2f;
typedef __attribute__((ext_vector_type(8))) float v8f;
// Exact pointee type of the async-to-LDS builtin's pointer params
// (from hipcc diagnostic: '__attribute__((__vector_size__(4*sizeof(int)))) int *')
typedef int v4i_g __attribute__((__vector_size__(16)));

#if defined(__gfx1250__) || defined(__AMDGCN__)
#if __has_builtin(__builtin_amdgcn_global_load_async_to_lds_b128)
#define XMI_ASYNC 1
#endif
#if __has_builtin(__builtin_amdgcn_wmma_f32_16x16x4_f32)
#define XMI_WMMA 1
#endif
#endif

// Full 32-lane sum, result broadcast to every lane.
// WMMA path: D = ones(16x4) x B(4x16) with B.x = v, B.y = 0 gives
// D[m,n] = v[n] + v[n+16] for every row m, so d[0] holds u[lane&15] in all
// lanes (robust to either 4x16 B-operand K-ordering). 4 xor-shuffles finish.
__device__ __forceinline__ float xmi_wave_sum(float v) {
#if defined(XMI_WMMA)
  v2f a; a.x = 1.0f; a.y = 1.0f;   // A = all ones (layout-independent)
  v2f b; b.x = v;    b.y = 0.0f;
  v8f c = {};
  c = __builtin_amdgcn_wmma_f32_16x16x4_f32(false, a, false, b,
                                            (short)0, c, false, false);
  float u = c[0];                  // u = v[lane&15] + v[(lane&15)+16]
  u += __shfl_xor(u, 1, 32);
  u += __shfl_xor(u, 2, 32);
  u += __shfl_xor(u, 4, 32);
  u += __shfl_xor(u, 8, 32);
  return u;
#else
  for (int off = 16; off; off >>= 1) v += __shfl_xor(v, off, 32);
  return v;
#endif
}

__device__ __forceinline__ float xmi_wave_max(float v) {
  for (int off = 16; off; off >>= 1) v = fmaxf(v, __shfl_xor(v, off, 32));
  return v;
}

// C (feature dim) is fixed at 256 by the reference problem.
#define XMI_C 256
#define XMI_ROWS_PER_BLOCK 8

__global__ __launch_bounds__(256) void xmi_stage1(
    const float* __restrict__ output,    // [N, K] log-prob matrix
    const int*   __restrict__ target,    // [N]
    const float* __restrict__ features,  // [N, 256]
    float*       __restrict__ part,      // [gridDim.x * 3] partials
    int K) {
  __shared__ __align__(16) float tile[XMI_ROWS_PER_BLOCK * XMI_C]; // 8 KB
  __shared__ float pt1[XMI_ROWS_PER_BLOCK];   // per-row  sum_j p*lf
  __shared__ float pt2[XMI_ROWS_PER_BLOCK];   // per-row  sum_j lf
  __shared__ float pnl[XMI_ROWS_PER_BLOCK];   // per-row  output[n, target[n]]

  const int tid  = threadIdx.x;
  const int wave = tid >> 5;
  const int lane = tid & 31;
  const long long rowBase = (long long)blockIdx.x * XMI_ROWS_PER_BLOCK;

  // ---- Stage the 8-row tile (8 KB) from global into LDS ----
  const char* gsrc = (const char*)(features + rowBase * XMI_C);
#if defined(XMI_ASYNC)
  {
    uintptr_t gbase = (uintptr_t)gsrc;
    uint32_t  lbase = (uint32_t)(uintptr_t)(void*)tile;  // low 32 bits == LDS byte addr
    const uint32_t o0 = (uint32_t)tid * 16u;             // 16B chunk per thread
    const uint32_t o1 = o0 + 4096u;                      // second half of tile
    __builtin_amdgcn_global_load_async_to_lds_b128(
        (AS1 v4i_g*)(gbase + o0), (AS3 v4i_g*)(lbase + o0), 0, 0);
    __builtin_amdgcn_global_load_async_to_lds_b128(
        (AS1 v4i_g*)(gbase + o1), (AS3 v4i_g*)(lbase + o1), 0, 0);
#if __has_builtin(__builtin_amdgcn_s_wait_asynccnt)
    __builtin_amdgcn_s_wait_asynccnt(0);
#else
    asm volatile("s_wait_asynccnt 0" ::: "memory");
#endif
  }
#else
  {
    const float4* g4 = (const float4*)gsrc;
    float4*       l4 = (float4*)tile;
    l4[tid]       = g4[tid];
    l4[tid + 256] = g4[tid + 256];
  }
#endif
  __syncthreads();

  // ---- Per-wave row: read 8 f32/lane from LDS (2x ds_load_b128) ----
  const float4* r4 = (const float4*)(tile + wave * XMI_C);
  float4 va = r4[lane];        // cols [4*lane .. 4*lane+3]
  float4 vb = r4[32 + lane];   // cols [128+4*lane .. 128+4*lane+3]

  // Row max
  float m = fmaxf(fmaxf(fmaxf(va.x, va.y), fmaxf(va.z, va.w)),
                  fmaxf(fmaxf(vb.x, vb.y), fmaxf(vb.z, vb.w)));
  m = xmi_wave_max(m);

  // Lane-local sums: se = sum exp(f-m), sf = sum f, sx = sum exp(f-m)*(f-m)
  float se = 0.0f, sf = 0.0f, sx = 0.0f;
  {
    float vals[8] = {va.x, va.y, va.z, va.w, vb.x, vb.y, vb.z, vb.w};
#pragma unroll
    for (int i = 0; i < 8; ++i) {
      float d = vals[i] - m;
      float e = __expf(d);
      se += e;
      sx += e * d;
      sf += vals[i];
    }
  }
  se = xmi_wave_sum(se);
  sf = xmi_wave_sum(sf);
  sx = xmi_wave_sum(sx);

  if (lane == 0) {
    float L = __logf(se);
    // sum_j p[j]*lf[j] = sx/se - L ;  sum_j lf[j] = sf - C*(m+L) ; sum_j p = 1
    pt1[wave] = sx / se - L;
    pt2[wave] = sf - (float)XMI_C * (m + L);
    long long n = rowBase + wave;
    pnl[wave] = output[n * (long long)K + (long long)target[n]];
  }
  __syncthreads();

  if (tid == 0) {
    float a = 0.0f, b = 0.0f, c = 0.0f;
#pragma unroll
    for (int w = 0; w < XMI_ROWS_PER_BLOCK; ++w) {
      a += pt1[w]; b += pt2[w]; c += pnl[w];
    }
    part[blockIdx.x * 3 + 0] = a;  // sum over rows of sum_j p*lf
    part[blockIdx.x * 3 + 1] = b;  // sum over rows of sum_j lf
    part[blockIdx.x * 3 + 2] = c;  // sum over rows of picked log-prob
  }
}

__global__ __launch_bounds__(256) void xmi_stage2(
    const float* __restrict__ part, float* __restrict__ out,
    int nparts, int N) {
  __shared__ double s0[256], s1[256], s2[256];
  const int tid = threadIdx.x;
  double a0 = 0.0, a1 = 0.0, a2 = 0.0;
  for (int i = tid; i < nparts; i += 256) {
    a0 += (double)part[3 * i + 0];
    a1 += (double)part[3 * i + 1];
    a2 += (double)part[3 * i + 2];
  }
  s0[tid] = a0; s1[tid] = a1; s2[tid] = a2;
  __syncthreads();
  for (int s = 128; s > 0; s >>= 1) {
    if (tid < s) {
      s0[tid] += s0[tid + s];
      s1[tid] += s1[tid + s];
      s2[tid] += s2[tid + s];
    }
    __syncthreads();
  }
  if (tid == 0) {
    const double inv_n = 1.0 / (double)N;
    double loss_nll = -s2[0] * inv_n;
    double loss_kl  = ((double)XMI_C * s0[0] - s1[0]) * inv_n;
    double loss     = 1.0 * loss_nll + 0.2 * loss_kl;
    out[0] = (float)loss;
    out[1] = (float)loss_nll;
    out[2] = (float)loss_kl;
  }
}

extern "C" void kernel_launch(void* const* d_in, const int* in_sizes, int n_in,
                              void* d_out, int out_size, void* d_ws, size_t ws_size,
                              hipStream_t stream) {
  const float* output   = (const float*)d_in[0];   // [N, K] f32
  const int*   target   = (const int*)d_in[1];     // [N] i32
  const float* features = (const float*)d_in[2];   // [N, 256] f32
  float* out  = (float*)d_out;                     // (loss, loss_nll, loss_kl)
  float* part = (float*)d_ws;                      // 3 floats per stage-1 block

  const int N = in_sizes[1];                       // 65536
  const int K = in_sizes[0] / N;                   // 1000
  const int nblocks = N / XMI_ROWS_PER_BLOCK;      // 8192 (N divisible by 8)

  xmi_stage1<<<nblocks, 256, 0, stream>>>(output, target, features, part, K);
  xmi_stage2<<<1, 256, 0, stream>>>(part, out, nblocks, N);
}